// CDR_SB_30803505447517
// MI455X (gfx1250) — compile-verified
//
#include <hip/hip_runtime.h>
#include <hip/hip_bf16.h>
#include <math.h>

typedef __attribute__((ext_vector_type(2))) float v2f;
typedef __attribute__((ext_vector_type(8))) float v8f;

#define DD 64
#define EPS_COS 1e-8f
#define EPS_BN  1e-5f

// ---------------------------------------------------------------------------
// Gather + cosine-sim gating. One 64-thread block per batch row.
// Dominant memory traffic: streams each history embedding row (256B) with
// float4 (global_load_b128) loads.
// ---------------------------------------------------------------------------
__global__ __launch_bounds__(64) void gather_gate_kernel(
    const float* __restrict__ embed_u_z, const float* __restrict__ embed_u_c,
    const float* __restrict__ embed_v_z, const float* __restrict__ embed_v_c,
    const float* __restrict__ item_table,
    const int* __restrict__ hist_items, const int* __restrict__ hist_lens,
    const int* __restrict__ item_ids, const int* __restrict__ red_by_friends,
    float* __restrict__ uv, int H) {
  __shared__ __align__(16) float anchor_s[DD];
  __shared__ float sim_s[64];
  __shared__ int   valid_s[64];

  const int b = blockIdx.x;
  const int t = threadIdx.x;
  const int item = item_ids[b];

  anchor_s[t] = item_table[(long)item * DD + t];
  __syncthreads();

  float sim = -INFINITY;
  int valid = 0;
  if (t < H) {
    const int hi = hist_items[b * H + t];
    const float4* he4 = (const float4*)(item_table + (long)hi * DD);
    const float4* an4 = (const float4*)anchor_s;
    float dot = 0.0f, na = 0.0f, nb = 0.0f;
#pragma unroll
    for (int i = 0; i < DD / 4; ++i) {
      float4 x = he4[i];
      float4 a = an4[i];
      dot += x.x * a.x + x.y * a.y + x.z * a.z + x.w * a.w;
      na  += x.x * x.x + x.y * x.y + x.z * x.z + x.w * x.w;
      nb  += a.x * a.x + a.y * a.y + a.z * a.z + a.w * a.w;
    }
    na = sqrtf(na);
    nb = sqrtf(nb);
    sim = dot / (fmaxf(na, EPS_COS) * fmaxf(nb, EPS_COS));
    valid = (t < hist_lens[b]) && (hi != item);
  }
  sim_s[t]   = valid ? sim : -INFINITY;
  valid_s[t] = valid;
  __syncthreads();

  for (int s = 32; s > 0; s >>= 1) {
    if (t < s) {
      sim_s[t]   = fmaxf(sim_s[t], sim_s[t + s]);
      valid_s[t] += valid_s[t + s];
    }
    __syncthreads();
  }

  const bool empty   = (valid_s[0] == 0);
  const bool gate    = (red_by_friends[b] > 0) && (sim_s[0] > 0.5f);
  const bool use_u_c = empty || gate;

  const long bd = (long)b * DD + t;
  const float u = embed_u_z[bd] + (use_u_c ? embed_u_c[bd] : 0.0f);
  // faithful to reference: empty-history v branch adds embed_u_c
  const float v = embed_v_z[bd] +
                  (empty ? embed_u_c[bd] : (gate ? embed_v_c[bd] : 0.0f));
  uv[(long)b * (2 * DD) + t]      = u;
  uv[(long)b * (2 * DD) + DD + t] = v;
}

// ---------------------------------------------------------------------------
// Generic MLP layer: y = relu( BN?(x) @ w + bias ).
// One wave (32 threads) per 16 output rows; WMMA F32 16x16x4 accumulation.
//
// A fragment (16x4 f32): lanes 0-15 -> row M=lane, K = k,k+1 ;
//                        lanes 16-31 -> row M=lane-16, K = k+2,k+3.
// B fragment (4x16 f32): VGPR0 rows K=k (lanes 0-15) / K=k+2 (lanes 16-31),
//                        VGPR1 rows K=k+1 / K=k+3 ; N = lane&15.
// C/D (16x16 f32): VGPR j -> M=j (lanes 0-15) / M=j+8 (lanes 16-31), N=lane&15.
// ---------------------------------------------------------------------------
__global__ __launch_bounds__(32) void mlp_layer_kernel(
    const float* __restrict__ x,       // [B, K]
    const float* __restrict__ w,       // [K, N]
    const float* __restrict__ bias,    // [N]
    const float* __restrict__ bn_sum,  // [K] (use_bn)
    const float* __restrict__ bn_sq,   // [K]
    const float* __restrict__ bn_g,    // [K]
    const float* __restrict__ bn_be,   // [K]
    float* __restrict__ y,             // [B, N]
    int K, int N, int use_bn, float invB) {
  __shared__ float sc_s[128];
  __shared__ float sh_s[128];

  const int lane = threadIdx.x;
  if (use_bn) {
    for (int c = lane; c < K; c += 32) {
      float mean = bn_sum[c] * invB;
      float var  = bn_sq[c] * invB - mean * mean;  // biased variance
      float sc   = bn_g[c] * rsqrtf(var + EPS_BN);
      sc_s[c] = sc;
      sh_s[c] = bn_be[c] - mean * sc;
    }
  } else {
    for (int c = lane; c < K; c += 32) {
      sc_s[c] = 1.0f;
      sh_s[c] = 0.0f;
    }
  }
  __syncthreads();

  const int row0  = blockIdx.x * 16;
  const int r     = lane & 15;
  const int khalf = (lane >> 4) * 2;
  const int col   = lane & 15;

  for (int nt = 0; nt < (N >> 4); ++nt) {
    v8f acc = {};
    for (int k = 0; k < K; k += 4) {
      const int kk = k + khalf;
      v2f a, bb;
      a.x  = sc_s[kk]     * x[(long)(row0 + r) * K + kk]     + sh_s[kk];
      a.y  = sc_s[kk + 1] * x[(long)(row0 + r) * K + kk + 1] + sh_s[kk + 1];
      bb.x = w[(long)kk * N + nt * 16 + col];
      bb.y = w[(long)(kk + 1) * N + nt * 16 + col];
      acc = __builtin_amdgcn_wmma_f32_16x16x4_f32(
          false, a, false, bb, (short)0, acc, false, false);
    }
    const int n = nt * 16 + col;
    const float bn = bias[n];
#pragma unroll
    for (int j = 0; j < 8; ++j) {
      const int m = row0 + j + (lane >> 4) * 8;
      y[(long)m * N + n] = fmaxf(acc[j] + bn, 0.0f);
    }
  }
}

// ---------------------------------------------------------------------------
// Deterministic per-column sum / sum-of-squares over y [B, N].
// One block per column; fixed LDS reduction tree -> bit-reproducible.
// ---------------------------------------------------------------------------
__global__ __launch_bounds__(256) void col_stats_kernel(
    const float* __restrict__ y, float* __restrict__ out_sum,
    float* __restrict__ out_sq, int Bn, int N) {
  __shared__ float ssum[256];
  __shared__ float ssq[256];
  const int n = blockIdx.x;
  const int t = threadIdx.x;
  float s = 0.0f, q = 0.0f;
  for (int m = t; m < Bn; m += 256) {
    float v = y[(long)m * N + n];
    s += v;
    q += v * v;
  }
  ssum[t] = s;
  ssq[t]  = q;
  __syncthreads();
  for (int st = 128; st > 0; st >>= 1) {
    if (t < st) {
      ssum[t] += ssum[t + st];
      ssq[t]  += ssq[t + st];
    }
    __syncthreads();
  }
  if (t == 0) {
    out_sum[n] = ssum[0];
    out_sq[n]  = ssq[0];
  }
}

// ---------------------------------------------------------------------------
// Final: BN3 then dot with wf[16] + bf
// ---------------------------------------------------------------------------
__global__ __launch_bounds__(256) void final_kernel(
    const float* __restrict__ y3, const float* __restrict__ sum3,
    const float* __restrict__ sq3, const float* __restrict__ g3,
    const float* __restrict__ be3, const float* __restrict__ wf,
    const float* __restrict__ bf, float* __restrict__ out, int Bn, float invB) {
  __shared__ float sc_s[16], sh_s[16], wf_s[16];
  const int t = threadIdx.x;
  if (t < 16) {
    float mean = sum3[t] * invB;
    float var  = sq3[t] * invB - mean * mean;
    float sc   = g3[t] * rsqrtf(var + EPS_BN);
    sc_s[t] = sc;
    sh_s[t] = be3[t] - mean * sc;
    wf_s[t] = wf[t];
  }
  __syncthreads();
  const int b = blockIdx.x * blockDim.x + t;
  if (b < Bn) {
    float s = 0.0f;
#pragma unroll
    for (int c = 0; c < 16; ++c)
      s += (sc_s[c] * y3[(long)b * 16 + c] + sh_s[c]) * wf_s[c];
    out[b] = s + bf[0];
  }
}

// ---------------------------------------------------------------------------
extern "C" void kernel_launch(void* const* d_in, const int* in_sizes, int n_in,
                              void* d_out, int out_size, void* d_ws,
                              size_t ws_size, hipStream_t stream) {
  const float* embed_u_z = (const float*)d_in[0];
  const float* embed_u_c = (const float*)d_in[1];
  const float* embed_v_z = (const float*)d_in[2];
  const float* embed_v_c = (const float*)d_in[3];
  const float* item_table = (const float*)d_in[4];
  const float* w1 = (const float*)d_in[5];
  const float* b1 = (const float*)d_in[6];
  const float* g1 = (const float*)d_in[7];
  const float* be1 = (const float*)d_in[8];
  const float* w2 = (const float*)d_in[9];
  const float* b2 = (const float*)d_in[10];
  const float* g2 = (const float*)d_in[11];
  const float* be2 = (const float*)d_in[12];
  const float* w3 = (const float*)d_in[13];
  const float* b3 = (const float*)d_in[14];
  const float* g3 = (const float*)d_in[15];
  const float* be3 = (const float*)d_in[16];
  const float* wf = (const float*)d_in[17];
  const float* bf = (const float*)d_in[18];
  const int* hist_items = (const int*)d_in[19];
  const int* hist_lens = (const int*)d_in[20];
  const int* item_ids = (const int*)d_in[21];
  const int* red_by_friends = (const int*)d_in[22];

  const int Bn = in_sizes[20];          // batch size
  const int H  = in_sizes[19] / Bn;     // history length
  const float invB = 1.0f / (float)Bn;

  float* ws = (float*)d_ws;
  float* uv = ws;                              // [B, 128]
  float* y1 = uv + (size_t)Bn * 128;           // [B, 64]
  float* y2 = y1 + (size_t)Bn * 64;            // [B, 32]
  float* y3 = y2 + (size_t)Bn * 32;            // [B, 16]
  float* stats = y3 + (size_t)Bn * 16;         // 224 floats
  float* sum1 = stats;      float* sq1 = sum1 + 64;
  float* sum2 = sq1 + 64;   float* sq2 = sum2 + 32;
  float* sum3 = sq2 + 32;   float* sq3 = sum3 + 16;

  gather_gate_kernel<<<Bn, 64, 0, stream>>>(
      embed_u_z, embed_u_c, embed_v_z, embed_v_c, item_table,
      hist_items, hist_lens, item_ids, red_by_friends, uv, H);

  // layer 1: [B,128] @ [128,64], no input BN
  mlp_layer_kernel<<<Bn / 16, 32, 0, stream>>>(
      uv, w1, b1, nullptr, nullptr, nullptr, nullptr,
      y1, 128, 64, 0, invB);
  col_stats_kernel<<<64, 256, 0, stream>>>(y1, sum1, sq1, Bn, 64);

  // layer 2: BN1 -> [B,64] @ [64,32]
  mlp_layer_kernel<<<Bn / 16, 32, 0, stream>>>(
      y1, w2, b2, sum1, sq1, g1, be1,
      y2, 64, 32, 1, invB);
  col_stats_kernel<<<32, 256, 0, stream>>>(y2, sum2, sq2, Bn, 32);

  // layer 3: BN2 -> [B,32] @ [32,16]
  mlp_layer_kernel<<<Bn / 16, 32, 0, stream>>>(
      y2, w3, b3, sum2, sq2, g2, be2,
      y3, 32, 16, 1, invB);
  col_stats_kernel<<<16, 256, 0, stream>>>(y3, sum3, sq3, Bn, 16);

  // final: BN3 -> @ wf + bf
  final_kernel<<<(Bn + 255) / 256, 256, 0, stream>>>(
      y3, sum3, sq3, g3, be3, wf, bf, (float*)d_out, Bn, invB);
}